// Mamba1_33371895890358
// MI455X (gfx1250) — compile-verified
//
#include <hip/hip_runtime.h>

// Problem constants (from reference)
#define BB   2
#define CIN  128
#define HH   32
#define WW0  48
#define HP   32     // token grid H'
#define WP   96     // token grid W' = 2*W
#define LSEQ 3072   // HP*WP
#define DI   256    // D_INNER
#define NST  16     // D_STATE
#define RNK  8      // DT_RANK
#define KD   4
#define NC   64     // scan chunks
#define CL   48     // chunk length (LSEQ/NC)

typedef float v2f __attribute__((ext_vector_type(2)));
typedef float v8f __attribute__((ext_vector_type(8)));

__device__ __forceinline__ float silu_f(float x) { return x / (1.0f + __expf(-x)); }
__device__ __forceinline__ float softplus_f(float x) {
  return (x > 20.0f) ? x : log1pf(__expf(x));
}
// hw-position of sequence index l for direction k
__device__ __forceinline__ int map_seq(int k, int l) {
  int s = (k >= 2) ? (LSEQ - 1 - l) : l;
  if (k & 1) { int hh = s & (HP - 1); int ww = s >> 5; return hh * WP + ww; }
  return s;
}

// ---------------- Kernel 1: token interleave + in_proj GEMM (WMMA f32) -------
__global__ __launch_bounds__(256) void k_inproj(
    const float* __restrict__ rgb, const float* __restrict__ tin,
    const float* __restrict__ w, float* __restrict__ xx, float* __restrict__ zb) {
  __shared__ float sx[16][CIN + 2];
  const int mblk = blockIdx.x;
  const int b  = mblk / (LSEQ / 16);
  const int l0 = (mblk % (LSEQ / 16)) * 16;
  const int c0 = blockIdx.y * 128;
  const int tid = threadIdx.x;
  for (int i = tid; i < 16 * CIN; i += 256) {
    int r = i >> 7, c = i & (CIN - 1);
    int l = l0 + r, hh = l / WP, ww = l % WP;
    const float* src = (ww & 1) ? tin : rgb;
    sx[r][c] = src[((b * CIN + c) * HH + hh) * WW0 + (ww >> 1)];
  }
  __syncthreads();
  const int wave = tid >> 5, lane = tid & 31;
  const int m = lane & 15, hi = lane >> 4, klo = hi << 1;
  const int n = c0 + wave * 16 + m;
  v8f acc = {};
  #pragma unroll 8
  for (int kb = 0; kb < CIN / 4; ++kb) {
    int kk = kb * 4 + klo;
    v2f a, bv;
    a.x = sx[m][kk];            a.y = sx[m][kk + 1];
    bv.x = w[n * CIN + kk];     bv.y = w[n * CIN + kk + 1];
    acc = __builtin_amdgcn_wmma_f32_16x16x4_f32(false, a, false, bv, 0, acc, false, false);
  }
  float* dst = (n < DI) ? (xx + (size_t)(b * DI + n) * LSEQ)
                        : (zb + (size_t)(b * DI + (n - DI)) * LSEQ);
  #pragma unroll
  for (int vr = 0; vr < 8; ++vr) dst[l0 + vr + (hi << 3)] = acc[vr];
}

// ---------------- Kernel 2: depthwise 3x3 conv + bias + SiLU -----------------
__global__ __launch_bounds__(256) void k_conv(
    const float* __restrict__ xx, const float* __restrict__ cw,
    const float* __restrict__ cb, float* __restrict__ xc) {
  int idx = blockIdx.x * 256 + threadIdx.x;
  if (idx >= BB * DI * HP * WP) return;
  int ww = idx % WP; int t1 = idx / WP;
  int hh = t1 % HP;  int t2 = t1 / HP;
  int d  = t2 % DI;  int b  = t2 / DI;
  const float* base = xx + (size_t)(b * DI + d) * LSEQ;
  float s = cb[d];
  #pragma unroll
  for (int dy = -1; dy <= 1; ++dy)
    #pragma unroll
    for (int dx = -1; dx <= 1; ++dx) {
      int y = hh + dy, x = ww + dx;
      if (y >= 0 && y < HP && x >= 0 && x < WP)
        s += cw[d * 9 + (dy + 1) * 3 + (dx + 1)] * base[y * WP + x];
    }
  xc[idx] = silu_f(s);
}

// ---------------- Kernel 3: x_proj GEMM + dt_proj GEMM + softplus (WMMA) -----
__global__ __launch_bounds__(128) void k_xproj(
    const float* __restrict__ xc, const float* __restrict__ xpw,
    const float* __restrict__ dtw, const float* __restrict__ dtb,
    float* __restrict__ Bs, float* __restrict__ Cs, float* __restrict__ dl) {
  __shared__ float sx[16][DI + 2];
  __shared__ float sd[16][48 + 2];
  const int gblk = blockIdx.x;
  const int b  = gblk / (LSEQ / 16);
  const int l0 = (gblk % (LSEQ / 16)) * 16;
  const int k  = blockIdx.y;
  const int tid = threadIdx.x;
  for (int i = tid; i < 16 * DI; i += 128) {
    int r = i >> 8, d = i & (DI - 1);
    int j = map_seq(k, l0 + r);
    sx[r][d] = xc[(size_t)(b * DI + d) * LSEQ + j];
  }
  __syncthreads();
  const int wave = tid >> 5, lane = tid & 31;
  const int m = lane & 15, hi = lane >> 4, klo = hi << 1;
  if (wave < 3) {                         // 3 n-tiles cover 40 (+8 pad) outputs
    const int n = wave * 16 + m;
    const bool valid = n < (RNK + 2 * NST);
    v8f acc = {};
    #pragma unroll 8
    for (int kb = 0; kb < DI / 4; ++kb) {
      int kk = kb * 4 + klo;
      v2f a, bv;
      a.x = sx[m][kk]; a.y = sx[m][kk + 1];
      bv.x = valid ? xpw[(k * 40 + n) * DI + kk]     : 0.0f;
      bv.y = valid ? xpw[(k * 40 + n) * DI + kk + 1] : 0.0f;
      acc = __builtin_amdgcn_wmma_f32_16x16x4_f32(false, a, false, bv, 0, acc, false, false);
    }
    #pragma unroll
    for (int vr = 0; vr < 8; ++vr) sd[vr + (hi << 3)][n] = acc[vr];
  }
  __syncthreads();
  for (int i = tid; i < 256; i += 128) {  // write Bs / Cs (b,k,l,n) contiguous
    int r = i >> 4, n = i & 15;
    size_t o = (size_t)((b * KD + k) * LSEQ + l0 + r) * NST + n;
    Bs[o] = sd[r][RNK + n];
    Cs[o] = sd[r][RNK + NST + n];
  }
  // delta = softplus(dts @ dt_w^T + bias), 16 d-tiles over 4 waves
  for (int t = wave; t < 16; t += 4) {
    int d0 = t * 16;
    v8f acc = {};
    #pragma unroll
    for (int kb = 0; kb < 2; ++kb) {
      int kk = kb * 4 + klo;
      v2f a, bv;
      a.x = sd[m][kk]; a.y = sd[m][kk + 1];
      bv.x = dtw[(k * DI + d0 + m) * RNK + kk];
      bv.y = dtw[(k * DI + d0 + m) * RNK + kk + 1];
      acc = __builtin_amdgcn_wmma_f32_16x16x4_f32(false, a, false, bv, 0, acc, false, false);
    }
    float bias = dtb[k * DI + d0 + m];
    #pragma unroll
    for (int vr = 0; vr < 8; ++vr) {
      int mm = vr + (hi << 3);
      dl[(size_t)((b * KD + k) * LSEQ + l0 + mm) * DI + d0 + m] =
          softplus_f(acc[vr] + bias);
    }
  }
}

// ------- Kernel 4a: chunked scan pass 1 — local states (h_in = 0) ------------
__global__ __launch_bounds__(256) void k_scan1(
    const float* __restrict__ xc, const float* __restrict__ dl,
    const float* __restrict__ Bs, const float* __restrict__ Alog,
    float* __restrict__ hloc, float* __restrict__ Ssum) {
  const int c  = blockIdx.x;        // chunk
  const int bk = blockIdx.y;        // b*KD + k
  const int b = bk >> 2, k = bk & 3;
  const int d = threadIdx.x;
  float Ar[NST], h[NST];
  #pragma unroll
  for (int n = 0; n < NST; ++n) {
    Ar[n] = -__expf(Alog[(k * DI + d) * NST + n]);
    h[n] = 0.0f;
  }
  const float* xcb = xc + (size_t)(b * DI + d) * LSEQ;
  const float* dlb = dl + (size_t)bk * LSEQ * DI + d;
  const float* Bb  = Bs + (size_t)bk * LSEQ * NST;
  float S = 0.0f;
  const int l0 = c * CL;
  for (int t = 0; t < CL; ++t) {
    int l = l0 + t;
    int j = map_seq(k, l);
    float dlt = dlb[(size_t)l * DI];
    float u = xcb[j];
    float du = dlt * u;
    S += dlt;
    const float4* bp = (const float4*)(Bb + l * NST);
    #pragma unroll
    for (int q = 0; q < 4; ++q) {
      float4 bq = bp[q];
      float bA[4] = {bq.x, bq.y, bq.z, bq.w};
      #pragma unroll
      for (int e = 0; e < 4; ++e) {
        int n = q * 4 + e;
        h[n] = __expf(dlt * Ar[n]) * h[n] + du * bA[e];
      }
    }
  }
  float4* hp = (float4*)(hloc + ((size_t)(bk * NC + c) * DI + d) * NST);
  #pragma unroll
  for (int q = 0; q < 4; ++q)
    hp[q] = make_float4(h[q * 4], h[q * 4 + 1], h[q * 4 + 2], h[q * 4 + 3]);
  Ssum[(size_t)(bk * NC + c) * DI + d] = S;
}

// ------- Kernel 4b: chunk-state combine (sequential over 64 chunks) ----------
// In-place: h_loc[c] becomes h_in[c] (incoming state at chunk start).
__global__ __launch_bounds__(256) void k_scan2(
    const float* __restrict__ Alog, float* hloc, const float* __restrict__ Ssum) {
  int idx = blockIdx.x * 256 + threadIdx.x;     // BB*KD*DI*NST = 32768
  int n = idx & (NST - 1), d = (idx >> 4) & (DI - 1), bk = idx >> 12;
  int k = bk & 3;
  float A = -__expf(Alog[(k * DI + d) * NST + n]);
  float hr = 0.0f;
  for (int c = 0; c < NC; ++c) {
    size_t o = ((size_t)(bk * NC + c) * DI + d) * NST + n;
    float hl = hloc[o];
    float S  = Ssum[(size_t)(bk * NC + c) * DI + d];
    hloc[o] = hr;                       // incoming state for chunk c
    hr = __expf(A * S) * hr + hl;       // exp(A*sum(delta)) == prod exp(A*delta)
  }
}

// ------- Kernel 4c: chunked scan pass 2 — full scan with y output ------------
__global__ __launch_bounds__(256) void k_scan3(
    const float* __restrict__ xc, const float* __restrict__ dl,
    const float* __restrict__ Bs, const float* __restrict__ Cs,
    const float* __restrict__ Alog, const float* __restrict__ Dvec,
    const float* __restrict__ hloc, float* __restrict__ y4) {
  const int c  = blockIdx.x;
  const int bk = blockIdx.y;
  const int b = bk >> 2, k = bk & 3;
  const int d = threadIdx.x;
  float Ar[NST], h[NST];
  #pragma unroll
  for (int n = 0; n < NST; ++n) Ar[n] = -__expf(Alog[(k * DI + d) * NST + n]);
  const float4* hp = (const float4*)(hloc + ((size_t)(bk * NC + c) * DI + d) * NST);
  #pragma unroll
  for (int q = 0; q < 4; ++q) {
    float4 hq = hp[q];
    h[q * 4] = hq.x; h[q * 4 + 1] = hq.y; h[q * 4 + 2] = hq.z; h[q * 4 + 3] = hq.w;
  }
  const float Dv = Dvec[k * DI + d];
  const float* xcb = xc + (size_t)(b * DI + d) * LSEQ;
  const float* dlb = dl + (size_t)bk * LSEQ * DI + d;
  const float* Bb  = Bs + (size_t)bk * LSEQ * NST;
  const float* Cb  = Cs + (size_t)bk * LSEQ * NST;
  float* yb = y4 + (size_t)((k * BB + b) * LSEQ) * DI + d;
  const int l0 = c * CL;
  for (int t = 0; t < CL; ++t) {
    int l = l0 + t;
    int j = map_seq(k, l);
    float dlt = dlb[(size_t)l * DI];
    float u = xcb[j];
    float du = dlt * u;
    const float4* bp = (const float4*)(Bb + l * NST);
    const float4* cp = (const float4*)(Cb + l * NST);
    float y = Dv * u;
    #pragma unroll
    for (int q = 0; q < 4; ++q) {
      float4 bq = bp[q], cq = cp[q];
      float bA[4] = {bq.x, bq.y, bq.z, bq.w};
      float cA[4] = {cq.x, cq.y, cq.z, cq.w};
      #pragma unroll
      for (int e = 0; e < 4; ++e) {
        int n = q * 4 + e;
        float dA = __expf(dlt * Ar[n]);
        h[n] = dA * h[n] + du * bA[e];
        y = fmaf(h[n], cA[e], y);
      }
    }
    yb[(size_t)j * DI] = y;   // scatter back to hw order
  }
}

// ---------------- Kernel 5: combine dirs + LN + SiLU gate + out_proj (WMMA) --
__global__ __launch_bounds__(256) void k_out(
    const float* __restrict__ y4, const float* __restrict__ zb,
    const float* __restrict__ gam, const float* __restrict__ bet,
    const float* __restrict__ pw, float* __restrict__ out) {
  __shared__ float sy[16][DI + 2];
  __shared__ float smu[16], srs[16];
  const int gblk = blockIdx.x;
  const int b  = gblk / 96;
  const int j0 = (gblk % 96) * 16;
  const int tid = threadIdx.x;
  for (int i = tid; i < 16 * DI; i += 256) {
    int r = i >> 8, d = i & (DI - 1);
    int l = 2 * (j0 + r) + 1;
    float acc = 0.0f;
    #pragma unroll
    for (int k = 0; k < KD; ++k)
      acc += y4[(size_t)((k * BB + b) * LSEQ + l) * DI + d];
    sy[r][d] = acc;
  }
  __syncthreads();
  if (tid < 16) {
    float s = 0.0f, s2 = 0.0f;
    for (int d = 0; d < DI; ++d) { float v = sy[tid][d]; s += v; s2 += v * v; }
    float mu = s / DI;
    smu[tid] = mu;
    srs[tid] = rsqrtf(fmaxf(s2 / DI - mu * mu, 0.0f) + 1e-5f);
  }
  __syncthreads();
  for (int i = tid; i < 16 * DI; i += 256) {
    int r = i >> 8, d = i & (DI - 1);
    int l = 2 * (j0 + r) + 1;
    float v = (sy[r][d] - smu[r]) * srs[r] * gam[d] + bet[d];
    float zz = zb[(size_t)(b * DI + d) * LSEQ + l];
    sy[r][d] = v * silu_f(zz);
  }
  __syncthreads();
  const int wave = tid >> 5, lane = tid & 31;
  const int m = lane & 15, hi = lane >> 4, klo = hi << 1;
  const int n = wave * 16 + m;  // 8 waves x 16 = 128 out channels
  v8f acc = {};
  #pragma unroll 8
  for (int kb = 0; kb < DI / 4; ++kb) {
    int kk = kb * 4 + klo;
    v2f a, bv;
    a.x = sy[m][kk];           a.y = sy[m][kk + 1];
    bv.x = pw[n * DI + kk];    bv.y = pw[n * DI + kk + 1];
    acc = __builtin_amdgcn_wmma_f32_16x16x4_f32(false, a, false, bv, 0, acc, false, false);
  }
  #pragma unroll
  for (int vr = 0; vr < 8; ++vr) {
    int j = j0 + vr + (hi << 3);
    int hh = j / WW0, wc = j % WW0;
    out[((size_t)(b * CIN + n) * HH + hh) * WW0 + wc] = acc[vr];
  }
}

extern "C" void kernel_launch(void* const* d_in, const int* in_sizes, int n_in,
                              void* d_out, int out_size, void* d_ws, size_t ws_size,
                              hipStream_t stream) {
  (void)in_sizes; (void)n_in; (void)out_size; (void)ws_size;
  const float* rgb        = (const float*)d_in[0];
  const float* tin        = (const float*)d_in[1];
  const float* in_proj_w  = (const float*)d_in[2];
  const float* conv_w     = (const float*)d_in[3];
  const float* conv_b     = (const float*)d_in[4];
  const float* x_proj_w   = (const float*)d_in[5];
  const float* dt_projs_w = (const float*)d_in[6];
  const float* dt_projs_b = (const float*)d_in[7];
  const float* A_logs     = (const float*)d_in[8];
  const float* Ds         = (const float*)d_in[9];
  const float* out_norm_g = (const float*)d_in[10];
  const float* out_norm_b = (const float*)d_in[11];
  const float* out_proj_w = (const float*)d_in[12];
  float* out = (float*)d_out;

  // Workspace layout (floats): ~81 MB total, fully overwritten each call.
  float* ws   = (float*)d_ws;
  float* xx   = ws;                                    // B*DI*L
  float* zb   = xx   + (size_t)BB * DI * LSEQ;         // B*DI*L
  float* xc   = zb   + (size_t)BB * DI * LSEQ;         // B*DI*L
  float* Bsb  = xc   + (size_t)BB * DI * LSEQ;         // B*K*L*N
  float* Csb  = Bsb  + (size_t)BB * KD * LSEQ * NST;   // B*K*L*N
  float* dlb  = Csb  + (size_t)BB * KD * LSEQ * NST;   // B*K*L*DI
  float* y4   = dlb  + (size_t)BB * KD * LSEQ * DI;    // K*B*L*DI
  float* hloc = y4   + (size_t)BB * KD * LSEQ * DI;    // B*K*NC*DI*NST
  float* Ssum = hloc + (size_t)BB * KD * NC * DI * NST;// B*K*NC*DI

  k_inproj<<<dim3(BB * LSEQ / 16, 4), 256, 0, stream>>>(rgb, tin, in_proj_w, xx, zb);
  k_conv<<<(BB * DI * HP * WP) / 256, 256, 0, stream>>>(xx, conv_w, conv_b, xc);
  k_xproj<<<dim3(BB * LSEQ / 16, KD), 128, 0, stream>>>(xc, x_proj_w, dt_projs_w,
                                                        dt_projs_b, Bsb, Csb, dlb);
  k_scan1<<<dim3(NC, BB * KD), DI, 0, stream>>>(xc, dlb, Bsb, A_logs, hloc, Ssum);
  k_scan2<<<(BB * KD * DI * NST) / 256, 256, 0, stream>>>(A_logs, hloc, Ssum);
  k_scan3<<<dim3(NC, BB * KD), DI, 0, stream>>>(xc, dlb, Bsb, Csb, A_logs, Ds, hloc, y4);
  k_out<<<BB * 96, 256, 0, stream>>>(y4, zb, out_norm_g, out_norm_b, out_proj_w, out);
}